// GfusedmaxList_76562087018944
// MI455X (gfx1250) — compile-verified
//
#include <hip/hip_runtime.h>
#include <hip/hip_bf16.h>
#include <stdint.h>

// Problem constants (match reference)
#define NG   4096      // graphs
#define MM   1024      // nodes per graph
#define EE   4096      // edges (shared across graphs)
#define N_ITERS 50
#define STEPSZ  0.2f
#define EPSV    1e-6f

typedef unsigned int v4u __attribute__((ext_vector_type(4)));
typedef int          v4i __attribute__((ext_vector_type(4)));
typedef int          v8i __attribute__((ext_vector_type(8)));

#if defined(__has_builtin)
#if __has_builtin(__builtin_amdgcn_tensor_load_to_lds)
#define HAVE_TDM 1
#endif
#endif

#ifdef HAVE_TDM
// Build a 1-D TDM descriptor (D#) and issue tensor_load_to_lds.
// n 4-byte elements from gptr -> LDS byte offset lds_off.
// D# group0: [1:0]=count=1, [63:32]=lds_addr, [120:64]=global_addr, [127:126]=type=2
// D# group1: [17:16]=data_size(2 -> 4B), [79:48]=tensor_dim0, [111:80]=tensor_dim1=1,
//            [127:112]=tile_dim0, [143:128]=tile_dim1=1, [207:160]=tensor_dim0_stride
__device__ __forceinline__ void tdm_load_1d(const void* gptr, unsigned lds_off, unsigned n) {
  unsigned long long ga = (unsigned long long)gptr;
  v4u g0;
  g0.x = 1u;                                   // count=1, user mode
  g0.y = lds_off;                              // lds_addr (bytes)
  g0.z = (unsigned)(ga & 0xFFFFFFFFull);       // global_addr[31:0]
  g0.w = (unsigned)((ga >> 32) & 0x01FFFFFFull) | (2u << 30); // addr[56:32] | type=2
  v8i g1;
  g1[0] = (int)(2u << 16);                     // data_size = 2 (4 bytes)
  g1[1] = (int)((n & 0xFFFFu) << 16);          // tensor_dim0[15:0] at bits[63:48]
  g1[2] = (int)(((n >> 16) & 0xFFFFu) | (1u << 16)); // tensor_dim0[31:16] | tensor_dim1=1
  g1[3] = (int)((n & 0xFFFFu) << 16);          // tile_dim0 = n at bits[127:112]
  g1[4] = 1;                                   // tile_dim1 = 1, tile_dim2 = 0
  g1[5] = (int)n;                              // tensor_dim0_stride low 32
  g1[6] = 0;                                   // stride highs
  g1[7] = 0;
  v4i z4 = {0, 0, 0, 0};
#if __has_include(<hip/amd_detail/amd_gfx1250_TDM.h>)
  v8i z8 = {0, 0, 0, 0, 0, 0, 0, 0};
  __builtin_amdgcn_tensor_load_to_lds(g0, g1, z4, z4, z8, 0); // clang-23 (6-arg)
#else
  __builtin_amdgcn_tensor_load_to_lds(g0, g1, z4, z4, 0);     // ROCm 7.2 (5-arg)
#endif
}
#endif

__global__ __launch_bounds__(MM) void gfusedmax_kernel(const float* __restrict__ x,
                                                       const int* __restrict__ edge,
                                                       float* __restrict__ out) {
  __shared__ float s_x[MM];
  __shared__ float s_y[MM];
  __shared__ float s_g[MM];
  __shared__ int2  s_edge[EE];       // interleaved (src,dst), read as 64-bit
  __shared__ float s_a[MM];          // sort buffer
  __shared__ float s_c[MM];          // scan buffer
  __shared__ float s_r0[MM];         // reduction: sum(zs*mask)
  __shared__ float s_r1[MM];         // reduction: sum(mask)

  const int tid  = threadIdx.x;
  const int b    = blockIdx.x;
  const long base = (long)b * MM;

#ifdef HAVE_TDM
  // One wave DMAs the row (4KB) and the edge list (32KB) into LDS via the
  // Tensor Data Mover; VALU stays free. Done-signal is per-issuing-wave, so
  // wave 0 waits on TENSORcnt, then the workgroup barrier publishes the LDS.
  if (tid < warpSize) {
    tdm_load_1d(x + base, (unsigned)(unsigned long long)(void*)s_x, MM);
    tdm_load_1d(edge,     (unsigned)(unsigned long long)(void*)s_edge, 2 * EE);
    __builtin_amdgcn_s_wait_tensorcnt(0);
  }
  __syncthreads();
#else
  s_x[tid] = x[base + tid];
  {
    const int* e32 = edge;
    #pragma unroll
    for (int i = tid; i < 2 * EE; i += MM) ((int*)s_edge)[i] = e32[i];
  }
  __syncthreads();
#endif

  // y0 = x; grad pre-initialized for iteration 0 (grad = y - x = 0)
  const float xv = s_x[tid];
  s_y[tid] = xv;
  s_g[tid] = 0.0f;
  __syncthreads();

  // ---- 50 iterations of graph fused lasso (lam = LAM/GAMMA = 1) ----
  // Phase A: edge scatter into grad (atomics). Phase B: y update fused with
  // next iteration's grad init (both thread-local) -> 2 barriers/iter.
  for (int it = 0; it < N_ITERS; ++it) {
    #pragma unroll
    for (int e = tid; e < EE; e += MM) {
      const int2 st = s_edge[e];           // one ds_load_b64
      const float d  = s_y[st.x] - s_y[st.y];
      const float gv = d * __builtin_amdgcn_rsqf(__builtin_fmaf(d, d, EPSV));
      atomicAdd(&s_g[st.x],  gv);          // ds_add_f32
      atomicAdd(&s_g[st.y], -gv);
    }
    __syncthreads();
    const float ynew = __builtin_fmaf(-STEPSZ, s_g[tid], s_y[tid]);
    s_y[tid] = ynew;
    s_g[tid] = ynew - xv;                  // grad init for next iteration
    __syncthreads();
  }

  // ---- sparsemax over the row ----
  // 1) bitonic sort descending (1024 elements, 1024 threads)
  s_a[tid] = s_y[tid];
  __syncthreads();
  for (int k = 2; k <= MM; k <<= 1) {
    for (int j = k >> 1; j > 0; j >>= 1) {
      const int ixj = tid ^ j;
      if (ixj > tid) {
        const float A = s_a[tid];
        const float B = s_a[ixj];
        const bool desc = ((tid & k) == 0);
        if ((A < B) == desc) { s_a[tid] = B; s_a[ixj] = A; }
      }
      __syncthreads();
    }
  }

  // 2) inclusive scan (cumsum) of sorted values, Hillis-Steele
  s_c[tid] = s_a[tid];
  __syncthreads();
  for (int off = 1; off < MM; off <<= 1) {
    float v = s_c[tid];
    if (tid >= off) v += s_c[tid - off];
    __syncthreads();
    s_c[tid] = v;
    __syncthreads();
  }

  // 3) support mask + two tree reductions -> tau
  {
    const float zs = s_a[tid];
    const float cs = s_c[tid];
    const float kk = (float)(tid + 1);
    const bool  m  = (1.0f + kk * zs) > cs;
    s_r0[tid] = m ? zs  : 0.0f;
    s_r1[tid] = m ? 1.0f : 0.0f;
  }
  __syncthreads();
  for (int s = MM >> 1; s > 0; s >>= 1) {
    if (tid < s) {
      s_r0[tid] += s_r0[tid + s];
      s_r1[tid] += s_r1[tid + s];
    }
    __syncthreads();
  }
  const float tau = (s_r0[0] - 1.0f) / s_r1[0];

  // 4) out = max(y - tau, 0)
  out[base + tid] = fmaxf(s_y[tid] - tau, 0.0f);
}

extern "C" void kernel_launch(void* const* d_in, const int* in_sizes, int n_in,
                              void* d_out, int out_size, void* d_ws, size_t ws_size,
                              hipStream_t stream) {
  (void)in_sizes; (void)n_in; (void)out_size; (void)d_ws; (void)ws_size;
  const float* x    = (const float*)d_in[0];
  const int*   edge = (const int*)d_in[1];  // (E,2) interleaved src,dst
  // d_in[2] (M_cumsum) is implied by the constant layout: row b starts at b*M.
  float* out = (float*)d_out;
  gfusedmax_kernel<<<NG, MM, 0, stream>>>(x, edge, out);
}